// BlockReLU_V1_39256001085537
// MI455X (gfx1250) — compile-verified
//
#include <hip/hip_runtime.h>

typedef __attribute__((ext_vector_type(4))) float v4f;

// activation: (N=16, C=128, H=128, W=128) fp32, row-major, W innermost.
// Each thread processes a 4x4 (H x W) tile of one channel:
//   4x global_load_b128 (one s_clause), register block-sums, 4x NT global_store_b128.
// Channel groups: c<32:(1,1)  c<64:(2,2)  c<96:(4,4)  c<112:(2,4)  else passthrough.
//
// Each 256-thread block lies entirely within one channel plane (1024 threads per
// (n,c) plane => 4 blocks per plane), so the group select is derived from
// blockIdx.x only -> SGPR-only value -> scalar s_cmp/s_cbranch, no EXEC masking.
__global__ __launch_bounds__(256) void block_relu_kernel(const float* __restrict__ in,
                                                         float* __restrict__ out) {
    const unsigned bid = blockIdx.x;
    const int nc  = bid >> 2;         // n*128 + c   (scalar, uniform)
    const int c   = nc & 127;         // scalar
    const int grp = c >> 5;           // scalar -> uniform branches below

    // Position within the 128x128 plane: plane_tid in [0,1024)
    const int ptid = ((bid & 3) << 8) | threadIdx.x;
    const int wt = ptid & 31;         // W-tile: lane-contiguous -> coalesced b128
    const int ht = ptid >> 5;         // H-tile: 0..31

    // base = nc*H*W + ht*4*W + wt*4  (floats); 16-byte aligned.
    const long base = ((long)nc << 14) + (ht << 9) + (wt << 2);

    const v4f* __restrict__ pin = reinterpret_cast<const v4f*>(in + base);
    // Row stride = 128 floats = 32 v4f.
    v4f r0 = pin[0];
    v4f r1 = pin[32];
    v4f r2 = pin[64];
    v4f r3 = pin[96];

    if (grp == 0) {
        // (1,1): per-element x * (x>=0)
        r0.x *= (r0.x >= 0.0f) ? 1.0f : 0.0f; r0.y *= (r0.y >= 0.0f) ? 1.0f : 0.0f;
        r0.z *= (r0.z >= 0.0f) ? 1.0f : 0.0f; r0.w *= (r0.w >= 0.0f) ? 1.0f : 0.0f;
        r1.x *= (r1.x >= 0.0f) ? 1.0f : 0.0f; r1.y *= (r1.y >= 0.0f) ? 1.0f : 0.0f;
        r1.z *= (r1.z >= 0.0f) ? 1.0f : 0.0f; r1.w *= (r1.w >= 0.0f) ? 1.0f : 0.0f;
        r2.x *= (r2.x >= 0.0f) ? 1.0f : 0.0f; r2.y *= (r2.y >= 0.0f) ? 1.0f : 0.0f;
        r2.z *= (r2.z >= 0.0f) ? 1.0f : 0.0f; r2.w *= (r2.w >= 0.0f) ? 1.0f : 0.0f;
        r3.x *= (r3.x >= 0.0f) ? 1.0f : 0.0f; r3.y *= (r3.y >= 0.0f) ? 1.0f : 0.0f;
        r3.z *= (r3.z >= 0.0f) ? 1.0f : 0.0f; r3.w *= (r3.w >= 0.0f) ? 1.0f : 0.0f;
    } else if (grp == 1) {
        // (2,2): four 2x2 sub-blocks of the 4x4 tile
        const float m00 = ((r0.x + r0.y + r1.x + r1.y) >= 0.0f) ? 1.0f : 0.0f;
        const float m01 = ((r0.z + r0.w + r1.z + r1.w) >= 0.0f) ? 1.0f : 0.0f;
        const float m10 = ((r2.x + r2.y + r3.x + r3.y) >= 0.0f) ? 1.0f : 0.0f;
        const float m11 = ((r2.z + r2.w + r3.z + r3.w) >= 0.0f) ? 1.0f : 0.0f;
        r0.x *= m00; r0.y *= m00; r1.x *= m00; r1.y *= m00;
        r0.z *= m01; r0.w *= m01; r1.z *= m01; r1.w *= m01;
        r2.x *= m10; r2.y *= m10; r3.x *= m10; r3.y *= m10;
        r2.z *= m11; r2.w *= m11; r3.z *= m11; r3.w *= m11;
    } else if (grp == 2) {
        // (4,4): whole tile is one block
        const float s = (r0.x + r0.y + r0.z + r0.w) + (r1.x + r1.y + r1.z + r1.w) +
                        (r2.x + r2.y + r2.z + r2.w) + (r3.x + r3.y + r3.z + r3.w);
        const float m = (s >= 0.0f) ? 1.0f : 0.0f;
        r0 *= m; r1 *= m; r2 *= m; r3 *= m;
    } else if (c < 112) {
        // (2,4): two 2x4 blocks (rows 0-1, rows 2-3)
        const float s0 = (r0.x + r0.y + r0.z + r0.w) + (r1.x + r1.y + r1.z + r1.w);
        const float s1 = (r2.x + r2.y + r2.z + r2.w) + (r3.x + r3.y + r3.z + r3.w);
        const float m0 = (s0 >= 0.0f) ? 1.0f : 0.0f;
        const float m1 = (s1 >= 0.0f) ? 1.0f : 0.0f;
        r0 *= m0; r1 *= m0;
        r2 *= m1; r3 *= m1;
    }
    // else c in [112,128): passthrough — store the loaded values unchanged.

    // Non-temporal stores: the 134MB output is never re-read; keep it from
    // evicting the (L2-resident, 134MB < 192MB) input across graph replays.
    v4f* __restrict__ pout = reinterpret_cast<v4f*>(out + base);
    __builtin_nontemporal_store(r0, pout);
    __builtin_nontemporal_store(r1, pout + 32);
    __builtin_nontemporal_store(r2, pout + 64);
    __builtin_nontemporal_store(r3, pout + 96);
}

extern "C" void kernel_launch(void* const* d_in, const int* in_sizes, int n_in,
                              void* d_out, int out_size, void* d_ws, size_t ws_size,
                              hipStream_t stream) {
    const float* act = (const float*)d_in[0];
    float* out = (float*)d_out;
    // 16*128*128*128 elements, 16 per thread -> 2,097,152 threads.
    const int total_threads = (16 * 128 * 128 * 128) / 16;
    const int block = 256;
    const int grid = total_threads / block;  // 8192
    block_relu_kernel<<<grid, block, 0, stream>>>(act, out);
}